// FPSsubsample_24867860644370
// MI455X (gfx1250) — compile-verified
//
#include <hip/hip_runtime.h>
#include <stddef.h>

// Problem constants (match reference)
#define BB 4
#define NN 2048
#define DD 3
#define CC 64
#define NS 512

typedef __attribute__((ext_vector_type(2))) float v2f;
typedef __attribute__((ext_vector_type(8))) float v8f;

// ---------------------------------------------------------------------------
// Threefry-2x32 (JAX PRNG) to reproduce far0 = randint(key(1), (4,), 0, 2048).
// key(1) -> (k0,k1) = (0,1). counts iota(4) split into x0=[0,1], x1=[2,3]:
// element b<2 uses pair (b, b+2) -> x0 word; b>=2 uses pair (b-2, b) -> x1 word.
// randint span = 2048 (power of two): offset = bits % 2048.
// ---------------------------------------------------------------------------
__device__ __forceinline__ unsigned rotl32(unsigned x, int r) {
  return (x << r) | (x >> (32 - r));
}

__device__ unsigned threefry_bits(int i) {
  const unsigned k0 = 0u, k1 = 1u;
  const unsigned k2 = 0x1BD11BDAu ^ k0 ^ k1;
  unsigned x0 = (i < 2) ? (unsigned)i : (unsigned)(i - 2);
  unsigned x1 = (i < 2) ? (unsigned)(i + 2) : (unsigned)i;
  const int ra[4] = {13, 15, 26, 6};
  const int rb[4] = {17, 29, 16, 24};
  x0 += k0; x1 += k1;
  #pragma unroll
  for (int r = 0; r < 4; ++r) { x0 += x1; x1 = rotl32(x1, ra[r]); x1 ^= x0; }
  x0 += k1; x1 += k2 + 1u;
  #pragma unroll
  for (int r = 0; r < 4; ++r) { x0 += x1; x1 = rotl32(x1, rb[r]); x1 ^= x0; }
  x0 += k2; x1 += k0 + 2u;
  #pragma unroll
  for (int r = 0; r < 4; ++r) { x0 += x1; x1 = rotl32(x1, ra[r]); x1 ^= x0; }
  x0 += k0; x1 += k1 + 3u;
  #pragma unroll
  for (int r = 0; r < 4; ++r) { x0 += x1; x1 = rotl32(x1, rb[r]); x1 ^= x0; }
  x0 += k1; x1 += k2 + 4u;
  #pragma unroll
  for (int r = 0; r < 4; ++r) { x0 += x1; x1 = rotl32(x1, ra[r]); x1 ^= x0; }
  x0 += k2; x1 += k0 + 5u;
  return (i < 2) ? x0 : x1;
}

// ---------------------------------------------------------------------------
// FPS kernel: one workgroup (1024 thr = 32 waves) per batch.
//
// Per step, the row of squared distances is computed on the matrix pipe via
// V_WMMA_F32_16X16X4_F32 (B = squared coords of 16 points, A = ones, so
// C[.,j] = sum_k coord_k(point j)^2; the K-sum is permutation invariant, so
// the exact B K-slot/lane mapping cannot affect the result).
//
// WMMA output mapping fixes point ownership: lane (wv, ln&15) owns points
// {wv*64 + g*16 + (ln&15) : g=0..3}, so the running min lives entirely in
// registers (dmin[4]) -- no LDS distance array, no DS RMW.
//
// Argmax uses a sortable 64-bit key: all candidates are non-negative floats,
// whose IEEE bits order monotonically as u32. key = (bits<<32) | ~index, so
// u64-max == (max value, then lowest index) -- jnp.argmax semantics, and the
// reduction is fully branchless (cmp_u64 + cndmask per shuffle step).
// ---------------------------------------------------------------------------
__global__ __launch_bounds__(1024)
void fps_kernel(const float* __restrict__ ab, const unsigned char* __restrict__ mask,
                float* __restrict__ out_mask, int* __restrict__ q_ws) {
  __shared__ int                qsh[NS];
  __shared__ unsigned long long redK[32];
  __shared__ int                farSh;

  const int b   = blockIdx.x;
  const int tid = threadIdx.x;
  const int wv  = tid >> 5;        // wave id 0..31
  const int ln  = tid & 31;        // lane id
  const int half = ln >> 4;        // 0: lanes 0-15, 1: lanes 16-31
  const int pl   = ln & 15;        // point-in-group

  float dmin[4] = {1e8f, 1e8f, 1e8f, 1e8f};   // running min, register-resident

  if (tid == 0) farSh = (int)(threefry_bits(b) & (NN - 1));  // % 2048
  __syncthreads();
  int far = farSh;

  const float wz = (half == 0) ? 1.0f : 0.0f;

  for (int s = 0; s < NS; ++s) {
    if (tid == 0) qsh[s] = far;

    const float* row = ab + ((size_t)b * NN + (size_t)far) * (size_t)(NN * DD);

    // --- d^2 of 64 points per wave: 8 clause-able loads, 4 WMMAs, 4 fmins --
    #pragma unroll
    for (int g = 0; g < 4; ++g) {
      const int p = wv * 64 + g * 16 + pl;  // this lane's point (dup per half)
      const float c0 = row[p * 3 + half];   // x (half 0) or y (half 1)
      const float c1 = row[p * 3 + 2];      // z (zeroed on half 1)
      v2f bm; bm[0] = c0 * c0; bm[1] = c1 * c1 * wz;
      v2f am; am[0] = 1.0f;    am[1] = 1.0f;
      v8f acc = {};
      acc = __builtin_amdgcn_wmma_f32_16x16x4_f32(
          /*neg_a=*/false, am, /*neg_b=*/false, bm,
          /*c_mod=*/(short)0, acc, /*reuse_a=*/false, /*reuse_b=*/false);
      dmin[g] = fminf(dmin[g], acc[0]);     // running min, no LDS
    }

    // --- local argmax over the 4 owned points (strict > keeps lowest g) ----
    float v  = dmin[0];
    int   ig = 0;
    #pragma unroll
    for (int g = 1; g < 4; ++g) {
      if (dmin[g] > v) { v = dmin[g]; ig = g; }
    }
    const int idx = wv * 64 + ig * 16 + pl;

    // --- branchless block argmax via sortable u64 key ----------------------
    unsigned long long key =
        ((unsigned long long)__float_as_uint(v) << 32) | (unsigned)(~idx);
    #pragma unroll
    for (int off = 16; off > 0; off >>= 1) {
      const unsigned long long ok = __shfl_xor(key, off, 32);
      key = (ok > key) ? ok : key;
    }
    if (ln == 0) redK[wv] = key;
    __syncthreads();
    if (wv == 0) {
      unsigned long long kk = redK[ln];
      #pragma unroll
      for (int off = 16; off > 0; off >>= 1) {
        const unsigned long long ok = __shfl_xor(kk, off, 32);
        kk = (ok > kk) ? ok : kk;
      }
      if (ln == 0) farSh = (int)(~(unsigned)kk);
    }
    __syncthreads();
    far = farSh;
  }

  // flush indices + gathered mask
  if (tid < NS) {
    const int qi = qsh[tid];
    q_ws[b * NS + tid] = qi;
    out_mask[b * NS + tid] = mask[b * NN + qi] ? 1.0f : 0.0f;
  }
}

// ---------------------------------------------------------------------------
// sub_ab[b,i,j,:] = ab[b, q[i], q[j], :]   (j fastest -> same source row per
// block stays hot in L2; writes fully contiguous)
// ---------------------------------------------------------------------------
__global__ __launch_bounds__(256)
void gather_ab_kernel(const float* __restrict__ ab, const int* __restrict__ q,
                      float* __restrict__ out_ab) {
  const int n = blockIdx.x * 256 + threadIdx.x;   // 0 .. B*NS*NS-1
  const int b = n >> 18;                          // / (512*512)
  const int r = n & 262143;
  const int i = r >> 9;
  const int j = r & 511;
  const int qi = q[b * NS + i];
  const int qj = q[b * NS + j];
  const size_t src = (((size_t)b * NN + (size_t)qi) * NN + (size_t)qj) * DD;
  const size_t dst = (size_t)n * DD;
  out_ab[dst + 0] = ab[src + 0];
  out_ab[dst + 1] = ab[src + 1];
  out_ab[dst + 2] = ab[src + 2];
}

// ---------------------------------------------------------------------------
// sub_vals[b,i,c] = values[b, q[i], c]  (coalesced 64-float rows)
// ---------------------------------------------------------------------------
__global__ __launch_bounds__(256)
void gather_vals_kernel(const float* __restrict__ vals, const int* __restrict__ q,
                        float* __restrict__ out_vals) {
  const int n = blockIdx.x * 256 + threadIdx.x;   // 0 .. B*NS*C-1
  const int b = n >> 15;                          // / (512*64)
  const int r = n & 32767;
  const int i = r >> 6;
  const int c = r & 63;
  const int qi = q[b * NS + i];
  out_vals[n] = vals[((size_t)b * NN + (size_t)qi) * CC + (size_t)c];
}

// ---------------------------------------------------------------------------
extern "C" void kernel_launch(void* const* d_in, const int* in_sizes, int n_in,
                              void* d_out, int out_size, void* d_ws, size_t ws_size,
                              hipStream_t stream) {
  (void)in_sizes; (void)n_in; (void)out_size; (void)ws_size;
  const float*         ab   = (const float*)d_in[0];          // (4,2048,2048,3) f32
  const float*         vals = (const float*)d_in[1];          // (4,2048,64) f32
  const unsigned char* mask = (const unsigned char*)d_in[2];  // (4,2048) bool

  float* out      = (float*)d_out;
  float* out_ab   = out;                                   // 4*512*512*3 = 3145728
  float* out_vals = out + (size_t)BB * NS * NS * DD;       // 4*512*64    = 131072
  float* out_mask = out_vals + (size_t)BB * NS * CC;       // 4*512       = 2048

  int* q_ws = (int*)d_ws;                                  // 4*512 ints

  fps_kernel<<<dim3(BB), dim3(1024), 0, stream>>>(ab, mask, out_mask, q_ws);

  const int n_ab = BB * NS * NS;                           // 1,048,576 threads
  gather_ab_kernel<<<dim3(n_ab / 256), dim3(256), 0, stream>>>(ab, q_ws, out_ab);

  const int n_v = BB * NS * CC;                            // 131,072 threads
  gather_vals_kernel<<<dim3(n_v / 256), dim3(256), 0, stream>>>(vals, q_ws, out_vals);
}